// RCNNTargetSampler_63926293233731
// MI455X (gfx1250) — compile-verified
//
#include <hip/hip_runtime.h>
#include <stdint.h>

// Problem constants (from reference)
#define NP      2000          // proposals per image
#define NG      100           // max GT per image
#define NT      2100          // NP + NG
#define NS      512           // samples out
#define MAXPOS  128           // NUM_SAMPLE * POS_RATIO
#define NNEG    384           // NS - MAXPOS
#define NIMG    128
#define BLK     256           // 8 wave32s
#define MAXM    9             // ceil(NT / BLK)

// ---- CDNA5 async global->LDS copy path (ASYNCcnt-tracked) -------------------
__device__ __forceinline__ uint32_t lds_off(const void* p) {
  // generic LDS pointer: low 32 bits are the LDS byte offset
  return (uint32_t)(uintptr_t)p;
}
__device__ __forceinline__ void async_ld_b128(uint32_t lds, uint64_t gaddr) {
  asm volatile("global_load_async_to_lds_b128 %0, %1, off"
               :: "v"(lds), "v"(gaddr) : "memory");
}
__device__ __forceinline__ void async_ld_b32(uint32_t lds, uint64_t gaddr) {
  asm volatile("global_load_async_to_lds_b32 %0, %1, off"
               :: "v"(lds), "v"(gaddr) : "memory");
}
__device__ __forceinline__ void wait_async0() {
  asm volatile("s_wait_asynccnt 0" ::: "memory");
}
// -----------------------------------------------------------------------------

__global__ __launch_bounds__(BLK)
void rcnn_target_sampler_kernel(const float* __restrict__ rois,
                                const float* __restrict__ scores,
                                const float* __restrict__ gtb,
                                const float* __restrict__ rnd,
                                float* __restrict__ out)
{
  __shared__ __align__(16) float s_gt[NG * 4];   // GT boxes (AoS rows)
  __shared__ float s_gtarea[NG];
  __shared__ float s_rand[NT];
  __shared__ int   s_cat[NT];                    // category by original index
  __shared__ int   s_arg[NT];                    // iou argmax by original index
  __shared__ int   s_perm[NT];                   // perm[k] = idx of k-th smallest rand
  __shared__ int   s_catp[NT];                   // category in rand order
  __shared__ int   s_cnt[4];                     // n3, n2, tail-n3, tail-n2

  const int tid = threadIdx.x;
  const int img = blockIdx.x;

  // --- 1. async DMA GT boxes (100 x float4) into LDS ---
  if (tid < NG) {
    async_ld_b128(lds_off(&s_gt[4 * tid]),
                  (uint64_t)(uintptr_t)(gtb + ((size_t)img * NG + tid) * 4));
  }
  wait_async0();
  __syncthreads();

  if (tid < 4) s_cnt[tid] = 0;
  if (tid < NG) {
    float w = s_gt[4 * tid + 2] - s_gt[4 * tid + 0];
    float h = s_gt[4 * tid + 3] - s_gt[4 * tid + 1];
    s_gtarea[tid] = fmaxf(w, 0.f) * fmaxf(h, 0.f);
  }

  // --- 2. kick off async copy of rand[] -> LDS; overlaps with IoU stage ---
  for (int i = tid; i < NT; i += BLK) {
    async_ld_b32(lds_off(&s_rand[i]),
                 (uint64_t)(uintptr_t)(rnd + (size_t)img * NT + i));
  }
  __syncthreads();   // publish s_gt areas / zeroed counters

  // --- 3. IoU max/argmax + category per candidate ---
  for (int i = tid; i < NT; i += BLK) {
    float x1, y1, x2, y2, sc;
    if (i < NP) {
      const float* p = rois + ((size_t)img * NP + i) * 4;
      x1 = p[0]; y1 = p[1]; x2 = p[2]; y2 = p[3];
      sc = scores[(size_t)img * NP + i];
    } else {
      int g = i - NP;
      x1 = s_gt[4 * g]; y1 = s_gt[4 * g + 1]; x2 = s_gt[4 * g + 2]; y2 = s_gt[4 * g + 3];
      sc = 1.0f;
    }
    float areaA = fmaxf(x2 - x1, 0.f) * fmaxf(y2 - y1, 0.f);
    float best = -1.f; int bi = 0;
    for (int j = 0; j < NG; ++j) {           // LDS broadcast reads (same j all lanes)
      float tx = fmaxf(x1, s_gt[4 * j + 0]);
      float ty = fmaxf(y1, s_gt[4 * j + 1]);
      float bx = fminf(x2, s_gt[4 * j + 2]);
      float by = fminf(y2, s_gt[4 * j + 3]);
      float w = fmaxf(bx - tx, 0.f), h = fmaxf(by - ty, 0.f);
      float inter = w * h;
      float uni = areaA + s_gtarea[j] - inter;
      float iou = (uni > 0.f) ? inter / uni : 0.f;
      if (iou > best) { best = iou; bi = j; }  // strict > keeps first max (jnp.argmax)
    }
    int c = 2;                 // default candidate-negative
    if (sc < 0.f)   c = 0;     // ignored
    if (best >= 0.5f) c = 3;   // positive (overrides ignore, as in reference)
    s_cat[i] = c; s_arg[i] = bi;
  }

  wait_async0();      // s_rand resident
  __syncthreads();    // + s_cat/s_arg published

  // --- 4. stable rank of rand (exact stable-argsort semantics) ---
  int myk[MAXM]; float myr[MAXM]; int rk[MAXM];
#pragma unroll
  for (int m = 0; m < MAXM; ++m) {
    int i = tid + m * BLK;
    myk[m] = (i < NT) ? i : -1;
    myr[m] = (i < NT) ? s_rand[i] : 0.f;
    rk[m] = 0;
  }
  for (int j = 0; j < NT; ++j) {
    float rj = s_rand[j];                      // LDS broadcast
#pragma unroll
    for (int m = 0; m < MAXM; ++m) {
      if (myk[m] >= 0)
        rk[m] += ((rj < myr[m]) || (rj == myr[m] && j < myk[m])) ? 1 : 0;
    }
  }
#pragma unroll
  for (int m = 0; m < MAXM; ++m)
    if (myk[m] >= 0) s_perm[rk[m]] = myk[m];   // ranks unique -> conflict-free scatter
  __syncthreads();

  // --- 5. categories in rand order + bucket counts ---
  for (int k = tid; k < NT; k += BLK) {
    int c = s_cat[s_perm[k]];
    s_catp[k] = c;
    if (c == 3)      { atomicAdd(&s_cnt[0], 1); if (k >= MAXPOS) atomicAdd(&s_cnt[2], 1); }
    else if (c == 2) { atomicAdd(&s_cnt[1], 1); if (k >= MAXPOS) atomicAdd(&s_cnt[3], 1); }
  }
  __syncthreads();
  const int n3  = s_cnt[0], n2  = s_cnt[1];
  const int m3t = s_cnt[2], m2t = s_cnt[3];

  // --- 6. stable bucket-sort positions + scatter outputs ---
  int cc[MAXM], c1[MAXM], c2[MAXM];
#pragma unroll
  for (int m = 0; m < MAXM; ++m) {
    int k = tid + m * BLK;
    cc[m] = (k < NT) ? s_catp[k] : -1;
    c1[m] = 0; c2[m] = 0;
  }
  for (int kp = 0; kp < NT; ++kp) {
    int cp   = s_catp[kp];                     // LDS broadcast
    int tail = (kp >= MAXPOS) ? 1 : 0;
#pragma unroll
    for (int m = 0; m < MAXM; ++m) {
      if (kp < myk[m] && cp == cc[m]) { c1[m]++; c2[m] += tail; }
    }
  }

  float* out_rois    = out;
  float* out_samples = out + (size_t)NIMG * NS * 4;
  float* out_matches = out + (size_t)NIMG * NS * 4 + (size_t)NIMG * NS;

#pragma unroll
  for (int m = 0; m < MAXM; ++m) {
    int k = myk[m];
    if (k < 0) continue;
    int c = cc[m];
    int i = s_perm[k];
    float smp = (c == 3) ? 1.f : ((c == 2) ? -1.f : 0.f);

    // gather the selected box once
    float b0, b1, b2, b3;
    if (i < NP) {
      const float* p = rois + ((size_t)img * NP + i) * 4;
      b0 = p[0]; b1 = p[1]; b2 = p[2]; b3 = p[3];
    } else {
      int g = i - NP;
      b0 = s_gt[4 * g]; b1 = s_gt[4 * g + 1]; b2 = s_gt[4 * g + 2]; b3 = s_gt[4 * g + 3];
    }
    float mch = (float)s_arg[i];

    // stage 1: descending stable sort of mask over all 2100 (3 > 2 > 0)
    int base1 = (c == 3) ? 0 : ((c == 2) ? n3 : (n3 + n2));
    int pos1  = base1 + c1[m];
    if (pos1 < MAXPOS) {
      size_t o = ((size_t)img * NS + pos1) * 4;
      out_rois[o + 0] = b0; out_rois[o + 1] = b1;
      out_rois[o + 2] = b2; out_rois[o + 3] = b3;
      out_samples[(size_t)img * NS + pos1] = smp;
      out_matches[(size_t)img * NS + pos1] = mch;
    }
    // stage 2: tail (k >= 128), mask 2 -> 4, descending (4 > 3 > 0)
    if (k >= MAXPOS) {
      int base2 = (c == 2) ? 0 : ((c == 3) ? m2t : (m2t + m3t));
      int pos2  = base2 + c2[m];
      if (pos2 < NNEG) {
        int slot = MAXPOS + pos2;
        size_t o = ((size_t)img * NS + slot) * 4;
        out_rois[o + 0] = b0; out_rois[o + 1] = b1;
        out_rois[o + 2] = b2; out_rois[o + 3] = b3;
        out_samples[(size_t)img * NS + slot] = smp;
        out_matches[(size_t)img * NS + slot] = mch;
      }
    }
  }
}

extern "C" void kernel_launch(void* const* d_in, const int* in_sizes, int n_in,
                              void* d_out, int out_size, void* d_ws, size_t ws_size,
                              hipStream_t stream) {
  const float* rois   = (const float*)d_in[0];   // (128, 2000, 4)
  const float* scores = (const float*)d_in[1];   // (128, 2000, 1)
  const float* gtb    = (const float*)d_in[2];   // (128, 100, 4)
  const float* rnd    = (const float*)d_in[3];   // (128, 2100)
  float* out = (float*)d_out;                    // rois | samples | matches, flat

  hipLaunchKernelGGL(rcnn_target_sampler_kernel,
                     dim3(NIMG), dim3(BLK), 0, stream,
                     rois, scores, gtb, rnd, out);
}